// RCNNBinDetLabelFromMatch_15719580304262
// MI455X (gfx1250) — compile-verified
//
#include <hip/hip_runtime.h>

// Problem constants (match reference)
#define B_      8
#define N_      512
#define NUM_GT_ 64
#define C_      16
#define FHW     16          // FH == FW == 16
#define TILE    (FHW*FHW)   // 256 elements per channel

typedef __attribute__((ext_vector_type(2))) float v2f;
typedef __attribute__((ext_vector_type(4))) float v4f;
typedef __attribute__((ext_vector_type(8))) float v8f;

#define LOG2E 1.44269504088896340736f

// One wave (32 lanes) per ROI. 8 waves per block.
__global__ __launch_bounds__(256)
void rcnn_bindet_encode_kernel(const float* __restrict__ boxes,      // (B,N,4)
                               const float* __restrict__ gt_boxes,   // (B,NUM_GT,5)
                               const int*   __restrict__ flag,       // (B,N)
                               const int*   __restrict__ gt_id,      // (B,N)
                               float*       __restrict__ out)        // label|offset|mask concat
{
    const int lane = threadIdx.x & 31;
    const int wave = threadIdx.x >> 5;
    // bn is identical across the wave: pin to SGPR so all uniform loads go SMEM.
    const int bn = __builtin_amdgcn_readfirstlane(blockIdx.x * 8 + wave);
    const int b  = bn / N_;

    // ---- uniform per-ROI loads (scalar path; no divergence before WMMA) ----
    const float bx1 = boxes[bn*4+0], by1 = boxes[bn*4+1];
    const float bx2 = boxes[bn*4+2], by2 = boxes[bn*4+3];
    const int   fl  = flag[bn];
    int g = gt_id[bn];
    g = (g < 0) ? 0 : ((g > NUM_GT_-1) ? (NUM_GT_-1) : g);   // clamp like _take_row
    const int grow = __builtin_amdgcn_readfirstlane((b*NUM_GT_ + g)*5);
    const float gx1 = gt_boxes[grow+0], gy1 = gt_boxes[grow+1];
    const float gx2 = gt_boxes[grow+2], gy2 = gt_boxes[grow+3];
    const float glab = gt_boxes[grow+4];

    // ---- anchor zoom (1.1x), relative gt, scales ----
    const float cx = (bx1+bx2)*0.5f, cy = (by1+by2)*0.5f;
    const float zw = (bx2-bx1)*1.1f, zh = (by2-by1)*1.1f;
    const float ax1 = cx - zw*0.5f,  ay1 = cy - zh*0.5f;
    const float ax2 = cx + zw*0.5f,  ay2 = cy + zh*0.5f;
    const float rx1 = gx1-ax1, ry1 = gy1-ay1, rx2 = gx2-ax1, ry2 = gy2-ay1;
    const float rw  = rx2-rx1, rh  = ry2-ry1;
    const float pcx = (rx1+rx2)*0.5f, pcy = (ry1+ry2)*0.5f;
    const float sw  = (ax2-ax1)*(1.0f/16.0f), sh = (ay2-ay1)*(1.0f/16.0f);
    const float inv_sw = 1.0f/sw, inv_sh = 1.0f/sh;
    const float w_sigma = rw*0.5f*inv_sw, h_sigma = rh*0.5f*inv_sh;
    const float w_sig_s = fmaxf(w_sigma, 1.5f), h_sig_s = fmaxf(h_sigma, 1.5f);
    const float w_sig_c = fmaxf(w_sigma, 0.5f), h_sig_c = fmaxf(h_sigma, 0.5f);
    const float pos_w = pcx*inv_sw, pos_h = pcy*inv_sh;
    const bool  small_ = (sw < 0.01f) || (sh < 0.01f);
    // class channel; labels are exact small ints so this is exact.
    const int   slab = __builtin_amdgcn_readfirstlane((int)fabsf(glab) - 1);

    // ---- separable heat factors on lanes 0..15 (masked by cond & small) ----
    const float fi = (float)(lane & 15) + 0.5f;          // ind + 0.5
    const float dw = (pos_w - fi) / w_sig_s;
    const float dh = (pos_h - fi) / h_sig_s;
    const bool  cw = fabsf(fi - pos_w) < w_sig_c;
    const bool  ch = fabsf(fi - pos_h) < h_sig_c;
    // branch-free native exp: exp(-x) = exp2(-x*log2e); arg <= 0 -> no range issues
    const float ew = __builtin_amdgcn_exp2f(-(dw*dw) * LOG2E);
    const float eh = __builtin_amdgcn_exp2f(-(dh*dh) * LOG2E);
    const float bvec = ((lane < 16) && cw)            ? ew : 0.0f;  // B[0][N=fw]
    const float avec = ((lane < 16) && ch && !small_) ? eh : 0.0f;  // A[M=fh][0]

    // ---- rank-1 outer product via f32 WMMA: heat = A(16x4,K0 only) * B(4x16,K0 only) ----
    v2f A;  A.x = avec;  A.y = 0.0f;    // lanes16-31 carry K=2,3 -> already zero via avec
    v2f Bm; Bm.x = bvec; Bm.y = 0.0f;   // only K=0 row (VGPR0, lanes 0-15) is non-zero
    v8f D = {};
    D = __builtin_amdgcn_wmma_f32_16x16x4_f32(false, A, false, Bm,
                                              (short)0, D, false, false);
    // D layout: D[r] at lane l -> heat[fh = r + 8*(l>=16)][fw = l&15]

    // ---- label_map stores: (bn, c, fh, fw) ----
    float* lbase = out + (size_t)bn * (C_*TILE);
    const v4f z4 = {0.f, 0.f, 0.f, 0.f};
    float* zb = lbase + lane*4;
    #pragma unroll
    for (int c = 0; c < C_; ++c) {
        if (c != slab) {                  // SCALAR branch (slab in SGPR)
            *(v4f*)(zb + c*TILE)       = z4;
            *(v4f*)(zb + c*TILE + 128) = z4;
        }
    }
    if (slab >= 0 && slab < C_) {         // scalar-uniform; full EXEC inside
        float* cb = lbase + slab*TILE + ((lane >> 4) << 7) + (lane & 15);
        #pragma unroll
        for (int r = 0; r < 8; ++r) cb[r*16] = D[r];
    }

    // ---- offset stores: (bn, ch, fh, fw); lane owns elements [lane*8, lane*8+8) ----
    // fh = lane>>1 (constant per lane); fw = (lane&1)*8 + j
    float* obase = out + (size_t)B_*N_*C_*TILE + (size_t)bn*4*TILE;
    const float fh_c  = (float)(lane >> 1) + 0.5f;
    const float basew = (float)((lane & 1) << 3) + 0.5f;
    const float vx1 = rx1*inv_sw, vx2 = rx2*inv_sw;
    const float oy1 = small_ ? 0.0f : (ry1*inv_sh - fh_c);
    const float oy2 = small_ ? 0.0f : (ry2*inv_sh - fh_c);
    v4f ox1a, ox1b, ox2a, ox2b;
    #pragma unroll
    for (int j = 0; j < 4; ++j) {
        const float t0 = basew + (float)j;
        const float t1 = basew + (float)(j + 4);
        ox1a[j] = small_ ? 0.0f : (vx1 - t0);
        ox1b[j] = small_ ? 0.0f : (vx1 - t1);
        ox2a[j] = small_ ? 0.0f : (vx2 - t0);
        ox2b[j] = small_ ? 0.0f : (vx2 - t1);
    }
    const v4f oy1v = {oy1, oy1, oy1, oy1};
    const v4f oy2v = {oy2, oy2, oy2, oy2};
    float* o0 = obase + lane*8;
    *(v4f*)(o0 + 0*TILE)     = ox1a;  *(v4f*)(o0 + 0*TILE + 4) = ox1b;
    *(v4f*)(o0 + 1*TILE)     = oy1v;  *(v4f*)(o0 + 1*TILE + 4) = oy1v;
    *(v4f*)(o0 + 2*TILE)     = ox2a;  *(v4f*)(o0 + 2*TILE + 4) = ox2b;
    *(v4f*)(o0 + 3*TILE)     = oy2v;  *(v4f*)(o0 + 3*TILE + 4) = oy2v;

    // ---- mask: (bn, c), float 0/1; CLS_ON_HARD == False ----
    float* mbase = out + (size_t)B_*N_*C_*TILE + (size_t)B_*N_*4*TILE + (size_t)bn*C_;
    const bool mask_one = (fl > 0) && (glab > 0.0f);
    if (lane < 16)
        mbase[lane] = ((lane == slab) && mask_one) ? 1.0f : 0.0f;
}

extern "C" void kernel_launch(void* const* d_in, const int* in_sizes, int n_in,
                              void* d_out, int out_size, void* d_ws, size_t ws_size,
                              hipStream_t stream) {
    const float* boxes    = (const float*)d_in[0];
    const float* gt_boxes = (const float*)d_in[1];
    const int*   flag     = (const int*)d_in[2];
    const int*   gt_id    = (const int*)d_in[3];
    float*       out      = (float*)d_out;

    const int n_roi  = B_ * N_;            // 4096 ROIs, 1 wave each
    const int blocks = n_roi / 8;          // 8 waves (256 thr) per block
    rcnn_bindet_encode_kernel<<<dim3(blocks), dim3(256), 0, stream>>>(
        boxes, gt_boxes, flag, gt_id, out);
}